// GAT_LSTM_22686017258134
// MI455X (gfx1250) — compile-verified
//
#include <hip/hip_runtime.h>
#include <hip/hip_bf16.h>
#include <math.h>

// ---------------- problem constants ----------------
#define NNODE 2000
#define NEDGE 16000
#define ETOT  18000          // edges + self loops
#define NG    40             // B*T graphs
#define MROWS 80000          // NG*NNODE (multiple of 16)
#define HHEAD 4
#define TSTEP 10
#define LDK   64000          // emb width = w_ih K dim
#define KCH   1600           // K chunk for lstm input gemm (64000/40)

typedef float v2f __attribute__((ext_vector_type(2)));
typedef float v8f __attribute__((ext_vector_type(8)));

static __device__ __forceinline__ void edge_sd(const int* __restrict__ ei,
                                               int e, int& s, int& d) {
  if (e < NEDGE) { s = ei[e]; d = ei[NEDGE + e]; }
  else           { s = e - NEDGE; d = e - NEDGE; }
}

// ============================================================
// fp32 WMMA GEMM: Out[M,Nn] = A[M,K] @ W[K,Nn], row-major.
// One wave per 16x16 C tile; M,Nn multiples of 16; K multiple of 4.
// ============================================================
__global__ void k_gemm_wmma(const float* __restrict__ A,
                            const float* __restrict__ W,
                            float* __restrict__ Out,
                            int M, int K, int Nn) {
  const int ntn  = Nn >> 4;
  const int tile = blockIdx.x;
  const int row0 = (tile / ntn) << 4;
  const int col0 = (tile % ntn) << 4;
  const int lane = threadIdx.x;
  const int half = lane >> 4;      // 0: K+{0,1}, 1: K+{2,3}
  const int l16  = lane & 15;
  const int m    = row0 + l16;

  v8f acc = {};
  for (int k = 0; k < K; k += 4) {
    const int kk = k + half * 2;
    v2f a, b;
    a.x = A[(size_t)m * K + kk];
    a.y = A[(size_t)m * K + kk + 1];
    b.x = W[(size_t)kk * Nn + col0 + l16];
    b.y = W[(size_t)(kk + 1) * Nn + col0 + l16];
    acc = __builtin_amdgcn_wmma_f32_16x16x4_f32(false, a, false, b,
                                                (short)0, acc, false, false);
  }
  for (int r = 0; r < 8; ++r) {
    const int mr = row0 + r + half * 8;
    Out[(size_t)mr * Nn + col0 + l16] = acc[r];
  }
}

// ============================================================
// LSTM input projection: G[40,1024] += emb[40,64000] @ w_ih[1024,64000]^T
// K split into 40 chunks of 1600; one wave per (n-tile, k-chunk);
// 3 M-tiles (rows 0..47, rows >=40 guarded) reuse the B operand.
// ============================================================
__global__ void k_lstm_gemm(const float* __restrict__ E,
                            const float* __restrict__ Wih,
                            float* __restrict__ G) {
  const int col0 = (blockIdx.x & 63) << 4;   // 64 tiles of N=1024
  const int k0   = (blockIdx.x >> 6) * KCH;  // 40 K chunks
  const int lane = threadIdx.x;
  const int half = lane >> 4;
  const int l16  = lane & 15;

  v8f acc0 = {}, acc1 = {}, acc2 = {};
  for (int k = k0; k < k0 + KCH; k += 4) {
    const int kk = k + half * 2;
    v2f b;
    b.x = Wih[(size_t)(col0 + l16) * LDK + kk];
    b.y = Wih[(size_t)(col0 + l16) * LDK + kk + 1];
    v2f a0, a1, a2;
    a0.x = E[(size_t)l16 * LDK + kk];
    a0.y = E[(size_t)l16 * LDK + kk + 1];
    a1.x = E[(size_t)(16 + l16) * LDK + kk];
    a1.y = E[(size_t)(16 + l16) * LDK + kk + 1];
    const int m2 = 32 + l16;
    a2.x = (m2 < NG) ? E[(size_t)m2 * LDK + kk] : 0.f;
    a2.y = (m2 < NG) ? E[(size_t)m2 * LDK + kk + 1] : 0.f;
    acc0 = __builtin_amdgcn_wmma_f32_16x16x4_f32(false, a0, false, b, (short)0, acc0, false, false);
    acc1 = __builtin_amdgcn_wmma_f32_16x16x4_f32(false, a1, false, b, (short)0, acc1, false, false);
    acc2 = __builtin_amdgcn_wmma_f32_16x16x4_f32(false, a2, false, b, (short)0, acc2, false, false);
  }
  const int n = col0 + l16;
  for (int r = 0; r < 8; ++r) {
    const int mA = r + half * 8;
    atomicAdd(&G[(size_t)mA * 1024 + n], acc0[r]);
    atomicAdd(&G[(size_t)(mA + 16) * 1024 + n], acc1[r]);
    const int m3 = mA + 32;
    if (m3 < NG) atomicAdd(&G[(size_t)m3 * 1024 + n], acc2[r]);
  }
}

// ================= CSR build (once per launch; graph shared by all 40) =====
__global__ void k_csr_zero(int* __restrict__ cnt, int* __restrict__ cur) {
  const int i = blockIdx.x * blockDim.x + threadIdx.x;
  if (i < NNODE) { cnt[i] = 0; cur[i] = 0; }
}
__global__ void k_csr_hist(const int* __restrict__ ei, int* __restrict__ cnt) {
  const int e = blockIdx.x * blockDim.x + threadIdx.x;
  if (e >= ETOT) return;
  int s, d; edge_sd(ei, e, s, d);
  atomicAdd(&cnt[d], 1);
}
// exclusive scan of cnt[0..1999] -> off[0..2000]; single block, 256 threads
__global__ void k_csr_scan(const int* __restrict__ cnt, int* __restrict__ off) {
  __shared__ int sp[256];
  const int t = threadIdx.x;
  int loc[8];
  int sum = 0;
  for (int j = 0; j < 8; ++j) {
    const int idx = t * 8 + j;
    const int v = (idx < NNODE) ? cnt[idx] : 0;
    loc[j] = sum;                 // exclusive within chunk
    sum += v;
  }
  sp[t] = sum; __syncthreads();
  for (int st = 1; st < 256; st <<= 1) {
    const int v = (t >= st) ? sp[t - st] : 0;
    __syncthreads();
    sp[t] += v;
    __syncthreads();
  }
  const int base = (t == 0) ? 0 : sp[t - 1];
  for (int j = 0; j < 8; ++j) {
    const int idx = t * 8 + j;
    if (idx < NNODE) off[idx] = base + loc[j];
  }
  if (t == 255) off[NNODE] = sp[255];
}
__global__ void k_csr_scatter(const int* __restrict__ ei, const int* __restrict__ off,
                              int* __restrict__ cur, int* __restrict__ eids) {
  const int e = blockIdx.x * blockDim.x + threadIdx.x;
  if (e >= ETOT) return;
  int s, d; edge_sd(ei, e, s, d);
  const int pos = off[d] + atomicAdd(&cur[d], 1);
  eids[pos] = e;
}

// ---------------- edge_attr mean over 16000 rows ----------------
__global__ void k_eamean(const float* __restrict__ ea, float* __restrict__ eam) {
  __shared__ float s0[256], s1[256];
  const int t = threadIdx.x;
  float a = 0.f, b = 0.f;
  for (int e = t; e < NEDGE; e += 256) { a += ea[e * 2]; b += ea[e * 2 + 1]; }
  s0[t] = a; s1[t] = b; __syncthreads();
  for (int st = 128; st > 0; st >>= 1) {
    if (t < st) { s0[t] += s0[t + st]; s1[t] += s1[t + st]; }
    __syncthreads();
  }
  if (t == 0) { eam[0] = s0[0] / NEDGE; eam[1] = s1[0] / NEDGE; }
}

// ---------------- ee[e,:] = ea_full(e) @ We  (K=2, shared by all graphs) ----
__global__ void k_ee(const float* __restrict__ ea, const float* __restrict__ eam,
                     const float* __restrict__ We, float* __restrict__ EE, int Fout) {
  const int i = blockIdx.x * blockDim.x + threadIdx.x;
  if (i >= ETOT * Fout) return;
  const int e = i / Fout, j = i - e * Fout;
  float a0, a1;
  if (e < NEDGE) { a0 = ea[e * 2]; a1 = ea[e * 2 + 1]; }
  else           { a0 = eam[0];    a1 = eam[1]; }
  EE[i] = a0 * We[j] + a1 * We[Fout + j];
}

// ============================================================
// GATv2 gather: one thread per (g, n, h). Two passes over the node's
// in-edges: (1) max logit, (2) p=exp(logit-max), denom and weighted sum
// of xl[src]. Epilogue fuses /denom, +bias, ReLU. No atomics.
// Fout = HHEAD*C; C compile-time so accumulators stay in registers.
// ============================================================
template <int C>
__global__ void k_gat_gather(const float* __restrict__ XL, const float* __restrict__ XR,
                             const float* __restrict__ EE, const float* __restrict__ att,
                             const float* __restrict__ bvec, const int* __restrict__ ei,
                             const int* __restrict__ off, const int* __restrict__ eids,
                             float* __restrict__ Out) {
  const int Fout = HHEAD * C;
  const int i = blockIdx.x * blockDim.x + threadIdx.x;
  if (i >= NG * NNODE * HHEAD) return;
  const int h  = i & 3;
  const int gn = i >> 2;
  const int g  = gn / NNODE;
  const int n  = gn - g * NNODE;
  const int e0 = off[n], e1 = off[n + 1];

  float xrv[C], av[C];
  {
    const float* xr = XR + ((size_t)g * NNODE + n) * Fout + h * C;
    const float* at = att + h * C;
    for (int c = 0; c < C; ++c) { xrv[c] = xr[c]; av[c] = at[c]; }
  }
  // pass 1: segment max of logits
  float mx = -INFINITY;
  for (int q = e0; q < e1; ++q) {
    const int e = eids[q];
    int s, d; edge_sd(ei, e, s, d);
    const float* xl = XL + ((size_t)g * NNODE + s) * Fout + h * C;
    const float* ep = EE + (size_t)e * Fout + h * C;
    float acc = 0.f;
    for (int c = 0; c < C; ++c) {
      float m = xl[c] + xrv[c] + ep[c];
      m = m > 0.f ? m : 0.2f * m;              // leaky_relu(0.2)
      acc += m * av[c];
    }
    mx = fmaxf(mx, acc);
  }
  // pass 2: exp-sum and weighted accumulation
  float den = 0.f;
  float vac[C];
  for (int c = 0; c < C; ++c) vac[c] = 0.f;
  for (int q = e0; q < e1; ++q) {
    const int e = eids[q];
    int s, d; edge_sd(ei, e, s, d);
    const float* xl = XL + ((size_t)g * NNODE + s) * Fout + h * C;
    const float* ep = EE + (size_t)e * Fout + h * C;
    float acc = 0.f;
    float xlv[C];
    for (int c = 0; c < C; ++c) {
      xlv[c] = xl[c];
      float m = xlv[c] + xrv[c] + ep[c];
      m = m > 0.f ? m : 0.2f * m;
      acc += m * av[c];
    }
    const float p = __expf(acc - mx);
    den += p;
    for (int c = 0; c < C; ++c) vac[c] += p * xlv[c];
  }
  const float inv = 1.f / (den + 1e-16f);
  float* o = Out + ((size_t)g * NNODE + n) * Fout + h * C;
  const float* bp = bvec + h * C;
  for (int c = 0; c < C; ++c) o[c] = fmaxf(vac[c] * inv + bp[c], 0.f);
}

// ---------------- G = b_ih + b_hh (broadcast); zero h/c state ----------------
__global__ void k_init_g(float* __restrict__ G, const float* __restrict__ bih,
                         const float* __restrict__ bhh, float* __restrict__ hc) {
  const int i = blockIdx.x * blockDim.x + threadIdx.x;
  if (i < NG * 1024) G[i] = bih[i & 1023] + bhh[i & 1023];
  if (i < 4096) hc[i] = 0.f;                   // 2 h-buffers + 2 c-buffers
}

// ---------------- one LSTM timestep (gates precomputed in G) ----------------
__global__ void k_lstm_step(const float* __restrict__ G, const float* __restrict__ Whh,
                            const float* __restrict__ hprev, const float* __restrict__ cprev,
                            float* __restrict__ hnext, float* __restrict__ cnext, int t) {
  const int b = blockIdx.x, j = threadIdx.x;   // 4 blocks x 256 threads
  const float* g = G + ((size_t)(b * TSTEP + t)) * 1024;
  float a0 = g[j], a1 = g[256 + j], a2 = g[512 + j], a3 = g[768 + j];
  const float* h = hprev + b * 256;
  for (int k = 0; k < 256; ++k) {
    const float hk = h[k];
    a0 += hk * Whh[(size_t)(      j) * 256 + k];
    a1 += hk * Whh[(size_t)(256 + j) * 256 + k];
    a2 += hk * Whh[(size_t)(512 + j) * 256 + k];
    a3 += hk * Whh[(size_t)(768 + j) * 256 + k];
  }
  const float ii = 1.f / (1.f + __expf(-a0));
  const float ff = 1.f / (1.f + __expf(-a1));
  const float gg = tanhf(a2);
  const float oo = 1.f / (1.f + __expf(-a3));
  const float c  = ff * cprev[b * 256 + j] + ii * gg;
  cnext[b * 256 + j] = c;
  hnext[b * 256 + j] = oo * tanhf(c);
}

// ---------------- head: relu -> fc1 -> relu -> fc2 ----------------
__global__ void k_fc(const float* __restrict__ hlast,
                     const float* __restrict__ fc1w, const float* __restrict__ fc1b,
                     const float* __restrict__ fc2w, const float* __restrict__ fc2b,
                     float* __restrict__ out) {
  __shared__ float s[512];
  const int b = blockIdx.x, j = threadIdx.x;   // 4 blocks x 512 threads
  float acc = fc1b[j];
  for (int k = 0; k < 256; ++k)
    acc += fmaxf(hlast[b * 256 + k], 0.f) * fc1w[(size_t)k * 512 + j];
  s[j] = fmaxf(acc, 0.f) * fc2w[j];
  __syncthreads();
  for (int st = 256; st > 0; st >>= 1) {
    if (j < st) s[j] += s[j + st];
    __syncthreads();
  }
  if (j == 0) out[b] = s[0] + fc2b[0];
}

// ============================================================
extern "C" void kernel_launch(void* const* d_in, const int* in_sizes, int n_in,
                              void* d_out, int out_size, void* d_ws, size_t ws_size,
                              hipStream_t stream) {
  const float* x    = (const float*)d_in[0];
  const int*   ei   = (const int*)  d_in[1];
  const float* ea   = (const float*)d_in[2];
  const float* Wl[3]  = {(const float*)d_in[3],  (const float*)d_in[8],  (const float*)d_in[13]};
  const float* Wr[3]  = {(const float*)d_in[4],  (const float*)d_in[9],  (const float*)d_in[14]};
  const float* We[3]  = {(const float*)d_in[5],  (const float*)d_in[10], (const float*)d_in[15]};
  const float* att[3] = {(const float*)d_in[6],  (const float*)d_in[11], (const float*)d_in[16]};
  const float* bb[3]  = {(const float*)d_in[7],  (const float*)d_in[12], (const float*)d_in[17]};
  const float* w_ih = (const float*)d_in[18];
  const float* w_hh = (const float*)d_in[19];
  const float* b_ih = (const float*)d_in[20];
  const float* b_hh = (const float*)d_in[21];
  const float* fc1w = (const float*)d_in[22];
  const float* fc1b = (const float*)d_in[23];
  const float* fc2w = (const float*)d_in[24];
  const float* fc2b = (const float*)d_in[25];

  // -------- workspace layout --------
  float* w = (float*)d_ws;
  size_t o = 0;
  float* XL  = w + o; o += (size_t)MROWS * 128;
  float* XR  = w + o; o += (size_t)MROWS * 128;
  float* HA  = w + o; o += (size_t)MROWS * 128;
  float* HB  = w + o; o += (size_t)MROWS * 128;
  float* EE  = w + o; o += (size_t)ETOT * 128;
  float* EAM = w + o; o += 8;
  float* G   = w + o; o += (size_t)NG * 1024;
  float* HS  = w + o; o += 2048;               // 2 h-buffers of [4,256]
  float* CS  = w + o; o += 2048;               // 2 c-buffers (contiguous after HS)
  int* CNT  = (int*)(w + o); o += 2048;
  int* OFF  = (int*)(w + o); o += 2048;        // 2001 used
  int* CUR  = (int*)(w + o); o += 2048;
  int* EIDS = (int*)(w + o); o += ETOT;
  (void)ws_size; (void)n_in; (void)in_sizes; (void)out_size;

  // ---- CSR of in-edges (topology shared by all 40 graphs) ----
  k_csr_zero<<<(NNODE + 255) / 256, 256, 0, stream>>>(CNT, CUR);
  k_csr_hist<<<(ETOT + 255) / 256, 256, 0, stream>>>(ei, CNT);
  k_csr_scan<<<1, 256, 0, stream>>>(CNT, OFF);
  k_csr_scatter<<<(ETOT + 255) / 256, 256, 0, stream>>>(ei, OFF, CUR, EIDS);
  k_eamean<<<1, 256, 0, stream>>>(ea, EAM);

  const int Fin_[3]  = {8, 128, 128};
  const int Fout_[3] = {128, 128, 32};
  const float* In_[3]  = {x, HA, HB};
  float*       Out_[3] = {HA, HB, HA};
  const int gth = NG * NNODE * HHEAD;          // 320,000 gather threads

  for (int l = 0; l < 3; ++l) {
    const int Fin = Fin_[l], Fout = Fout_[l];
    const int tiles = (MROWS / 16) * (Fout / 16);
    k_gemm_wmma<<<tiles, 32, 0, stream>>>(In_[l], Wl[l], XL, MROWS, Fin, Fout);
    k_gemm_wmma<<<tiles, 32, 0, stream>>>(In_[l], Wr[l], XR, MROWS, Fin, Fout);
    k_ee<<<(ETOT * Fout + 255) / 256, 256, 0, stream>>>(ea, EAM, We[l], EE, Fout);
    if (l < 2)
      k_gat_gather<32><<<(gth + 255) / 256, 256, 0, stream>>>(XL, XR, EE, att[l], bb[l],
                                                              ei, OFF, EIDS, Out_[l]);
    else
      k_gat_gather<8><<<(gth + 255) / 256, 256, 0, stream>>>(XL, XR, EE, att[l], bb[l],
                                                             ei, OFF, EIDS, Out_[l]);
  }

  // emb = HA viewed as [40, 64000] (node-major per graph == reshape(-1))
  k_init_g<<<(NG * 1024 + 255) / 256, 256, 0, stream>>>(G, b_ih, b_hh, HS);
  k_lstm_gemm<<<64 * 40, 32, 0, stream>>>(HA, w_ih, G);

  for (int t = 0; t < TSTEP; ++t) {
    float* hp = HS + (t & 1) * 1024;
    float* cp = CS + (t & 1) * 1024;
    float* hn = HS + ((t + 1) & 1) * 1024;
    float* cn = CS + ((t + 1) & 1) * 1024;
    k_lstm_step<<<4, 256, 0, stream>>>(G, w_hh, hp, cp, hn, cn, t);
  }
  // after t=9 the final h lives in buffer 0
  k_fc<<<4, 512, 0, stream>>>(HS, fc1w, fc1b, fc2w, fc2b, (float*)d_out);
}